// GraphQNNAutoencoder_65481071400988
// MI455X (gfx1250) — compile-verified
//
#include <hip/hip_runtime.h>
#include <hip/hip_bf16.h>
#include <math.h>

// ---------------------------------------------------------------------------
// Sizes (fixed by the reference)
// ---------------------------------------------------------------------------
#define NN     1024
#define F_IN   512
#define F_MID  768
#define F_OUT  1024
#define FLATSZ (F_OUT * F_OUT)   // 1048576
#define H1SZ   128
#define H2SZ   64
#define LATSZ  32
#define KCHUNKS 64               // split-K chunks for the We1 GEMV

typedef __attribute__((ext_vector_type(2))) float v2f;
typedef __attribute__((ext_vector_type(8))) float v8f;

// Fragment set for one k-step (K advance of 4): one A fragment + 4 B fragments
// (the wave owns a 16x64 tile => 4 N-subtiles reuse the same A fragment).
struct Frag {
    v2f a, b0, b1, b2, b3;
};

// ---------------------------------------------------------------------------
// fp32 WMMA GEMM: C[M,N] = act(A[M,K] @ opB + bias)
//   TRANSB=1: opB(k,n) = W[n*K + k]   (torch Linear weight [N,K] row-major)
//   TRANSB=0: opB(k,n) = B[k*N + n]   (plain row-major B)
// One wave computes a 16x64 tile with 4 accumulators. The k-loop is manually
// software-pipelined (double-buffered fragments) so the loads for stage k+8
// are issued before the WMMAs consuming stage k -- avoids the
// load -> s_wait_loadcnt 0 -> wmma serialization seen in the previous build.
// blockDim.x == 32 so EXEC is all ones at every WMMA (ISA 7.12 requirement).
// ---------------------------------------------------------------------------
template <bool TRANSB, bool TANH>
__global__ __launch_bounds__(32)
void wmma_gemm_f32(const float* __restrict__ A, const float* __restrict__ B,
                   const float* __restrict__ bias, float* __restrict__ C,
                   int M, int N, int K) {
    const int lane  = threadIdx.x;       // 0..31
    const int h     = lane >> 4;         // half-wave select (K interleave)
    const int ln    = lane & 15;         // M index (A) / N index (B, C)
    const int ntile = blockIdx.x * 64;   // 4 consecutive 16-wide N tiles
    const int mtile = blockIdx.y * 16;

    // A fragment source: A[mtile+ln][k0 + 2h .. +1]  (contiguous float2)
    const float* arow = A + (size_t)(mtile + ln) * K + 2 * h;
    // B fragment source (TRANSB): W[ntile+16t+ln][k0 + 2h .. +1] (float2)
    const float* brow = B + (size_t)(ntile + ln) * K + 2 * h;

    auto load_frag = [&](int k0) -> Frag {
        Frag f;
        f.a = *(const v2f*)(arow + k0);
        if (TRANSB) {
            f.b0 = *(const v2f*)(brow + k0);
            f.b1 = *(const v2f*)(brow + (size_t)16 * K + k0);
            f.b2 = *(const v2f*)(brow + (size_t)32 * K + k0);
            f.b3 = *(const v2f*)(brow + (size_t)48 * K + k0);
        } else {
            const float* bp = B + (size_t)(k0 + 2 * h) * N + ntile + ln;
            f.b0[0] = bp[0];  f.b0[1] = bp[N];
            f.b1[0] = bp[16]; f.b1[1] = bp[N + 16];
            f.b2[0] = bp[32]; f.b2[1] = bp[N + 32];
            f.b3[0] = bp[48]; f.b3[1] = bp[N + 48];
        }
        return f;
    };

    v8f acc0 = {}, acc1 = {}, acc2 = {}, acc3 = {};
    auto do_wmma = [&](const Frag& f) {
        // 8 args: (neg_a, A, neg_b, B, c_mod, C, reuse_a, reuse_b)
        acc0 = __builtin_amdgcn_wmma_f32_16x16x4_f32(false, f.a, false, f.b0,
                                                     (short)0, acc0, false, false);
        acc1 = __builtin_amdgcn_wmma_f32_16x16x4_f32(false, f.a, false, f.b1,
                                                     (short)0, acc1, false, false);
        acc2 = __builtin_amdgcn_wmma_f32_16x16x4_f32(false, f.a, false, f.b2,
                                                     (short)0, acc2, false, false);
        acc3 = __builtin_amdgcn_wmma_f32_16x16x4_f32(false, f.a, false, f.b3,
                                                     (short)0, acc3, false, false);
    };

    // Depth-2 software pipeline over K (K is always a multiple of 8 here).
    Frag f0 = load_frag(0);
    Frag f1 = load_frag(4);
    for (int k0 = 0; k0 + 8 < K; k0 += 4) {
        Frag fn = load_frag(k0 + 8);   // issue next-next stage loads first
        do_wmma(f0);                   // consume oldest stage
        f0 = f1;
        f1 = fn;
    }
    do_wmma(f0);
    do_wmma(f1);

    // Epilogue: C/D layout: VGPR v, lane l -> row = v + 8h, col = l&15.
    v8f accs[4] = {acc0, acc1, acc2, acc3};
#pragma unroll
    for (int t = 0; t < 4; ++t) {
        const int ncol = ntile + 16 * t + ln;
        const float bs = bias ? bias[ncol] : 0.0f;
#pragma unroll
        for (int v = 0; v < 8; ++v) {
            const int row = mtile + v + 8 * h;
            float x = accs[t][v] + bs;
            if (TANH) x = tanhf(x);
            C[(size_t)row * N + ncol] = x;
        }
    }
}

// ---------------------------------------------------------------------------
// Split-K GEMV partials: part[row][chunk] = dot(We1[row, chunk*CH : +CH], x[...])
// 512 MB of We1 streamed once with float4 loads; deterministic (no atomics).
// ---------------------------------------------------------------------------
__global__ __launch_bounds__(256)
void gemv_we1_partial(const float* __restrict__ We1, const float* __restrict__ x,
                      float* __restrict__ part) {
    const int CH   = FLATSZ / KCHUNKS;    // 16384 elements per chunk
    const int row  = blockIdx.x;          // 0..127
    const int chnk = blockIdx.y;          // 0..63
    const float4* w  = (const float4*)(We1 + (size_t)row * FLATSZ + (size_t)chnk * CH);
    const float4* xv = (const float4*)(x + (size_t)chnk * CH);

    float s = 0.0f;
    for (int i = threadIdx.x; i < CH / 4; i += 256) {
        __builtin_prefetch(w + i + 512, 0, 0);   // global_prefetch_b8, stream ahead
        const float4 a = w[i];
        const float4 b = xv[i];
        s += a.x * b.x + a.y * b.y + a.z * b.z + a.w * b.w;
    }
    __shared__ float red[256];
    red[threadIdx.x] = s;
    __syncthreads();
    for (int off = 128; off > 0; off >>= 1) {
        if (threadIdx.x < off) red[threadIdx.x] += red[threadIdx.x + off];
        __syncthreads();
    }
    if (threadIdx.x == 0) part[row * KCHUNKS + chnk] = red[0];
}

// ---------------------------------------------------------------------------
// Tiny MLP (deterministic serial reductions; trivial FLOPs). One block.
//   z1 = relu(sum(part) + be1); z2 = relu(We2 z1 + be2); z3 = We3 z2 + be3
//   y1 = relu(Wd1 z3 + bd1);    y2 = relu(Wd2 y1 + bd2)  -> y2out[128]
// ---------------------------------------------------------------------------
__global__ __launch_bounds__(128)
void mlp_small(const float* __restrict__ part, const float* __restrict__ be1,
               const float* __restrict__ We2, const float* __restrict__ be2,
               const float* __restrict__ We3, const float* __restrict__ be3,
               const float* __restrict__ Wd1, const float* __restrict__ bd1,
               const float* __restrict__ Wd2, const float* __restrict__ bd2,
               float* __restrict__ y2out) {
    __shared__ float z1[H1SZ], z2[H2SZ], z3[LATSZ], y1[H2SZ];
    const int t = threadIdx.x;

    if (t < H1SZ) {
        float s = be1[t];
        for (int c = 0; c < KCHUNKS; ++c) s += part[t * KCHUNKS + c];
        z1[t] = fmaxf(s, 0.0f);
    }
    __syncthreads();
    if (t < H2SZ) {
        float s = be2[t];
        for (int k = 0; k < H1SZ; ++k) s += We2[t * H1SZ + k] * z1[k];
        z2[t] = fmaxf(s, 0.0f);
    }
    __syncthreads();
    if (t < LATSZ) {
        float s = be3[t];
        for (int k = 0; k < H2SZ; ++k) s += We3[t * H2SZ + k] * z2[k];
        z3[t] = s;
    }
    __syncthreads();
    if (t < H2SZ) {
        float s = bd1[t];
        for (int k = 0; k < LATSZ; ++k) s += Wd1[t * LATSZ + k] * z3[k];
        y1[t] = fmaxf(s, 0.0f);
    }
    __syncthreads();
    if (t < H1SZ) {
        float s = bd2[t];
        for (int k = 0; k < H2SZ; ++k) s += Wd2[t * H2SZ + k] * y1[k];
        y2out[t] = fmaxf(s, 0.0f);
    }
}

// ---------------------------------------------------------------------------
// Final GEMV: out[i] = dot(Wd3[i,0:128], y2) + bd3[i], i in [0, 1M).
// One wave reads one full 128-float row per global_load_b128 (lane l takes
// floats 4l..4l+3 -> fully coalesced 512B), butterfly-reduces on wave32,
// then stores 32 consecutive outputs coalesced. Streams 512 MB of Wd3 once.
// ---------------------------------------------------------------------------
__global__ __launch_bounds__(256)
void gemv_wd3(const float* __restrict__ Wd3, const float* __restrict__ y2,
              const float* __restrict__ bd3, float* __restrict__ out) {
    const int lane   = threadIdx.x & 31;
    const int wave   = (blockIdx.x * blockDim.x + threadIdx.x) >> 5;
    const int nwaves = (gridDim.x * blockDim.x) >> 5;

    const float4 yv = ((const float4*)y2)[lane];  // lane owns y2[4l..4l+3]

    for (int rbase = wave * 32; rbase < FLATSZ; rbase += nwaves * 32) {
        float res = 0.0f;
#pragma unroll 4
        for (int r = 0; r < 32; ++r) {
            const float4* wp = (const float4*)(Wd3 + (size_t)(rbase + r) * H1SZ);
            __builtin_prefetch(wp + 32 * 8, 0, 0);   // 8 rows ahead in the stream
            const float4 w = wp[lane];
            float s = w.x * yv.x + w.y * yv.y + w.z * yv.z + w.w * yv.w;
            for (int off = 16; off > 0; off >>= 1) s += __shfl_xor(s, off, 32);
            if (lane == r) res = s;
        }
        out[rbase + lane] = res + bd3[rbase + lane];
    }
}

// ---------------------------------------------------------------------------
// Launch
// ---------------------------------------------------------------------------
extern "C" void kernel_launch(void* const* d_in, const int* in_sizes, int n_in,
                              void* d_out, int out_size, void* d_ws, size_t ws_size,
                              hipStream_t stream) {
    const float* X   = (const float*)d_in[0];
    const float* Adj = (const float*)d_in[1];
    const float* W1  = (const float*)d_in[2];
    const float* b1  = (const float*)d_in[3];
    const float* W2  = (const float*)d_in[4];
    const float* b2  = (const float*)d_in[5];
    const float* We1 = (const float*)d_in[6];
    const float* be1 = (const float*)d_in[7];
    const float* We2 = (const float*)d_in[8];
    const float* be2 = (const float*)d_in[9];
    const float* We3 = (const float*)d_in[10];
    const float* be3 = (const float*)d_in[11];
    const float* Wd1 = (const float*)d_in[12];
    const float* bd1 = (const float*)d_in[13];
    const float* Wd2 = (const float*)d_in[14];
    const float* bd2 = (const float*)d_in[15];
    const float* Wd3 = (const float*)d_in[16];
    const float* bd3 = (const float*)d_in[17];

    float* ws   = (float*)d_ws;
    float* H1   = ws;                         // 1024*768
    float* H2   = H1 + (size_t)NN * F_MID;    // 1024*768
    float* H3   = H2 + (size_t)NN * F_MID;    // 1024*1024
    float* H4   = H3 + (size_t)NN * F_OUT;    // 1024*1024 (== flat)
    float* part = H4 + (size_t)NN * F_OUT;    // 128*64 split-K partials
    float* y2   = part + H1SZ * KCHUNKS;      // 128

    // GNN: h1 = tanh(X @ W1^T + b1)
    wmma_gemm_f32<true, true><<<dim3(F_MID / 64, NN / 16), 32, 0, stream>>>(
        X, W1, b1, H1, NN, F_MID, F_IN);
    // h2 = Adj @ h1
    wmma_gemm_f32<false, false><<<dim3(F_MID / 64, NN / 16), 32, 0, stream>>>(
        Adj, H1, nullptr, H2, NN, F_MID, NN);
    // h3 = tanh(h2 @ W2^T + b2)
    wmma_gemm_f32<true, true><<<dim3(F_OUT / 64, NN / 16), 32, 0, stream>>>(
        H2, W2, b2, H3, NN, F_OUT, F_MID);
    // h4 = Adj @ h3  (flattened == H4)
    wmma_gemm_f32<false, false><<<dim3(F_OUT / 64, NN / 16), 32, 0, stream>>>(
        Adj, H3, nullptr, H4, NN, F_OUT, NN);

    // Encoder first layer (512 MB stream): split-K partials, then tiny MLP.
    gemv_we1_partial<<<dim3(H1SZ, KCHUNKS), 256, 0, stream>>>(We1, H4, part);
    mlp_small<<<1, 128, 0, stream>>>(part, be1, We2, be2, We3, be3,
                                     Wd1, bd1, Wd2, bd2, y2);

    // Decoder last layer (512 MB stream): out = Wd3 @ y2 + bd3.
    gemv_wd3<<<1024, 256, 0, stream>>>(Wd3, y2, bd3, (float*)d_out);
}